// PAU_5892695130631
// MI455X (gfx1250) — compile-verified
//
#include <hip/hip_runtime.h>
#include <stdint.h>

typedef float v4f __attribute__((ext_vector_type(4)));

// ---------------------------------------------------------------------------
// PAU: out = P(z)/Q(z), z = x + c
//   P(z) = a0 + a1 z + ... + a5 z^5                        (Horner, 5 FMA)
//   Q(z) = 1 + |z|(|b1| + |z|(|b2| + |z|(|b3| + |z||b4|))) (Horner, 4 FMA)
// HBM-bound streaming kernel: b128 NT loads/stores (GVS addressing, 32-bit
// voffset), rcp + 1 Newton step for the divide (Q >= 1 always).
// ---------------------------------------------------------------------------

__device__ __forceinline__ float pau_eval(float xv, float c,
                                          float a0, float a1, float a2,
                                          float a3, float a4, float a5,
                                          float b1, float b2, float b3, float b4) {
    float z  = xv + c;
    float az = __builtin_fabsf(z);
    float p = __builtin_fmaf(a5, z, a4);
    p = __builtin_fmaf(p, z, a3);
    p = __builtin_fmaf(p, z, a2);
    p = __builtin_fmaf(p, z, a1);
    p = __builtin_fmaf(p, z, a0);
    float q = __builtin_fmaf(b4, az, b3);
    q = __builtin_fmaf(q, az, b2);
    q = __builtin_fmaf(q, az, b1);
    q = __builtin_fmaf(q, az, 1.0f);
    float r = __builtin_amdgcn_rcpf(q);      // Q in [1, inf) => well conditioned
    float e = __builtin_fmaf(-q, r, 1.0f);
    r = __builtin_fmaf(r, e, r);
    return p * r;
}

#define PAU_COEFFS_LOAD                                       \
    const float c  = center[0];                               \
    const float a0 = numer[0], a1 = numer[1], a2 = numer[2];  \
    const float a3 = numer[3], a4 = numer[4], a5 = numer[5];  \
    const float b1 = __builtin_fabsf(denom[0]);               \
    const float b2 = __builtin_fabsf(denom[1]);               \
    const float b3 = __builtin_fabsf(denom[2]);               \
    const float b4 = __builtin_fabsf(denom[3]);

// Main kernel: every block processes exactly 4*256 float4 groups (4096 floats,
// 16 KB in / 16 KB out). NO bounds checks -> no EXEC predication, loads issue
// as an unbroken clause. 32-bit byte offsets -> SGPR-base + VGPR-offset (GVS)
// addressing. Only launched for the fully-covered prefix.
__global__ __launch_bounds__(256) void pau_main32_kernel(
    const float* __restrict__ x,
    const float* __restrict__ center,
    const float* __restrict__ numer,
    const float* __restrict__ denom,
    float* __restrict__ out)
{
    PAU_COEFFS_LOAD

    // byte offset of this thread's first float4 group
    const uint32_t off0 = blockIdx.x * (4u * 256u * 16u) + threadIdx.x * 16u;
    const uint32_t step = 256u * 16u;   // one block-stride of float4 groups

    const char* __restrict__ xb = (const char*)x;
    char* __restrict__ ob = (char*)out;

    v4f v0 = __builtin_nontemporal_load((const v4f*)(xb + off0));
    v4f v1 = __builtin_nontemporal_load((const v4f*)(xb + off0 + step));
    v4f v2 = __builtin_nontemporal_load((const v4f*)(xb + off0 + 2u * step));
    v4f v3 = __builtin_nontemporal_load((const v4f*)(xb + off0 + 3u * step));

    v4f r0, r1, r2, r3;
#pragma unroll
    for (int k = 0; k < 4; ++k) {
        r0[k] = pau_eval(v0[k], c, a0, a1, a2, a3, a4, a5, b1, b2, b3, b4);
        r1[k] = pau_eval(v1[k], c, a0, a1, a2, a3, a4, a5, b1, b2, b3, b4);
        r2[k] = pau_eval(v2[k], c, a0, a1, a2, a3, a4, a5, b1, b2, b3, b4);
        r3[k] = pau_eval(v3[k], c, a0, a1, a2, a3, a4, a5, b1, b2, b3, b4);
    }

    __builtin_nontemporal_store(r0, (v4f*)(ob + off0));
    __builtin_nontemporal_store(r1, (v4f*)(ob + off0 + step));
    __builtin_nontemporal_store(r2, (v4f*)(ob + off0 + 2u * step));
    __builtin_nontemporal_store(r3, (v4f*)(ob + off0 + 3u * step));
}

// 64-bit-offset variant (only used if buffers exceed 4 GB; not hit for the
// reference shape). Same structure, no bounds checks.
__global__ __launch_bounds__(256) void pau_main64_kernel(
    const float* __restrict__ x,
    const float* __restrict__ center,
    const float* __restrict__ numer,
    const float* __restrict__ denom,
    float* __restrict__ out)
{
    PAU_COEFFS_LOAD

    const v4f* __restrict__ xin = (const v4f*)x;
    v4f* __restrict__ o = (v4f*)out;
    const long long base = (long long)blockIdx.x * (4LL * 256LL) + threadIdx.x;

    v4f v0 = __builtin_nontemporal_load(&xin[base]);
    v4f v1 = __builtin_nontemporal_load(&xin[base + 256]);
    v4f v2 = __builtin_nontemporal_load(&xin[base + 512]);
    v4f v3 = __builtin_nontemporal_load(&xin[base + 768]);

    v4f r0, r1, r2, r3;
#pragma unroll
    for (int k = 0; k < 4; ++k) {
        r0[k] = pau_eval(v0[k], c, a0, a1, a2, a3, a4, a5, b1, b2, b3, b4);
        r1[k] = pau_eval(v1[k], c, a0, a1, a2, a3, a4, a5, b1, b2, b3, b4);
        r2[k] = pau_eval(v2[k], c, a0, a1, a2, a3, a4, a5, b1, b2, b3, b4);
        r3[k] = pau_eval(v3[k], c, a0, a1, a2, a3, a4, a5, b1, b2, b3, b4);
    }

    __builtin_nontemporal_store(r0, &o[base]);
    __builtin_nontemporal_store(r1, &o[base + 256]);
    __builtin_nontemporal_store(r2, &o[base + 512]);
    __builtin_nontemporal_store(r3, &o[base + 768]);
}

// Edge kernel: scalar, bounds-checked, covers [start, n). Only launched when
// n is not an exact multiple of 4096 (never for the reference shape).
__global__ __launch_bounds__(256) void pau_edge_kernel(
    const float* __restrict__ x,
    const float* __restrict__ center,
    const float* __restrict__ numer,
    const float* __restrict__ denom,
    float* __restrict__ out,
    long long start, long long n)
{
    PAU_COEFFS_LOAD

    long long i = start + (long long)blockIdx.x * blockDim.x + threadIdx.x;
    if (i < n) {
        out[i] = pau_eval(x[i], c, a0, a1, a2, a3, a4, a5, b1, b2, b3, b4);
    }
}

extern "C" void kernel_launch(void* const* d_in, const int* in_sizes, int n_in,
                              void* d_out, int out_size, void* d_ws, size_t ws_size,
                              hipStream_t stream) {
    (void)n_in; (void)out_size; (void)d_ws; (void)ws_size;

    const float* x      = (const float*)d_in[0];
    const float* center = (const float*)d_in[1];
    const float* numer  = (const float*)d_in[2];
    const float* denom  = (const float*)d_in[3];
    float* out = (float*)d_out;

    const long long n = (long long)in_sizes[0];   // 134,217,728 for ref shape
    const long long ELEMS_PER_BLOCK = 4096;       // 4 * 256 float4 groups

    const long long fullBlocks = n / ELEMS_PER_BLOCK;   // 32768 for ref shape
    const long long covered    = fullBlocks * ELEMS_PER_BLOCK;

    if (fullBlocks > 0) {
        // 32-bit byte offsets are valid while max offset < 4 GB.
        if (covered * 4 <= 0xFFFFFFFFLL) {
            pau_main32_kernel<<<(unsigned)fullBlocks, 256, 0, stream>>>(
                x, center, numer, denom, out);
        } else {
            pau_main64_kernel<<<(unsigned)fullBlocks, 256, 0, stream>>>(
                x, center, numer, denom, out);
        }
    }

    if (covered < n) {
        const long long rem = n - covered;                 // < 4096
        const unsigned eblocks = (unsigned)((rem + 255) / 256);
        pau_edge_kernel<<<eblocks, 256, 0, stream>>>(
            x, center, numer, denom, out, covered, n);
    }
}